// BiMamba2DFast_18124761989165
// MI455X (gfx1250) — compile-verified
//
#include <hip/hip_runtime.h>
#include <hip/hip_bf16.h>

// ---------------------------------------------------------------------------
// BiMamba2D (Vision-Mamba style) for gfx1250 / MI455X.
// All GEMMs use V_WMMA_F32_16X16X4_F32 (full-precision f32 matrix path).
// ---------------------------------------------------------------------------

#define D_MODEL 192
#define D_STATE 16
#define D_INNER 384
#define DT_RANK 12
#define XPROJ_N (DT_RANK + 2 * D_STATE)   // 44
#define NSEQ    112                        // B*W = B*H = 2*56
#define SEQL    56
#define NTOK    (NSEQ * SEQL)              // 6272  (multiple of 16)

typedef float v2f __attribute__((ext_vector_type(2)));
typedef float v8f __attribute__((ext_vector_type(8)));

// ---------------------------------------------------------------------------
// Generic f32 WMMA GEMM:  D[M x N] = A[M x K] * W[N x K]^T.
// MASKED: compile-time flag for N not a multiple of 16 (x_proj, N=44->48).
// OP:     0 = plain store, 1 = softplus(v + bias[col]) (dt projection).
// One wave32 computes one 16x16 output tile.
//
// A layout (16x4 f32, ISA 7.12.2): lane L: m = L&15, k = kbase + 2*(L>>4)+v
// B layout (4x16):                 lane L: n = L&15, k = kbase + 2*(L>>4)+v
//                                  (W row n supplies column n of B = W^T)
// C/D layout: lane L: n = L&15, row = j + 8*(L>>4) for accumulator VGPR j.
// ---------------------------------------------------------------------------
template <bool MASKED, int OP>
__global__ __launch_bounds__(256)
void wmma_gemm_f32(const float* __restrict__ A, int ldA,
                   const float* __restrict__ W, int ldW, int Nreal,
                   const float* __restrict__ bias,
                   float* __restrict__ D, int ldD,
                   int tilesM, int tilesN, int K)
{
    int wave = (blockIdx.x * (int)blockDim.x + (int)threadIdx.x) >> 5;
    int lane = threadIdx.x & 31;
    int total = tilesM * tilesN;
    if (wave >= total) return;                 // wave-uniform exit: EXEC all-1s

    int tm = wave / tilesN;
    int tn = wave - tm * tilesN;
    int mrow = tm * 16 + (lane & 15);          // A row for this lane
    int ecol = tn * 16 + (lane & 15);          // output column / weight row
    int khalf = (lane >> 4) * 2;               // K sub-offset per lane half

    int   eC    = ecol;
    float emask = 1.0f;
    if (MASKED) {
        eC    = (ecol < Nreal) ? ecol : (Nreal - 1);
        emask = (ecol < Nreal) ? 1.0f : 0.0f;
    }

    const float* arow = A + (size_t)mrow * ldA + khalf;
    const float* wrow = W + (size_t)eC   * ldW + khalf;

    v8f acc = {0.f, 0.f, 0.f, 0.f, 0.f, 0.f, 0.f, 0.f};
    for (int k = 0; k < K; k += 4) {
        v2f a, b;
        a[0] = arow[k];
        a[1] = arow[k + 1];
        if (MASKED) {
            b[0] = wrow[k] * emask;
            b[1] = wrow[k + 1] * emask;
        } else {
            b[0] = wrow[k];
            b[1] = wrow[k + 1];
        }
        acc = __builtin_amdgcn_wmma_f32_16x16x4_f32(
            /*neg_a=*/false, a, /*neg_b=*/false, b,
            /*c_mod=*/(short)0, acc, /*reuse_a=*/false, /*reuse_b=*/false);
    }

    bool colOK = MASKED ? (ecol < Nreal) : true;
    int  rbase = tm * 16 + 8 * (lane >> 4);
    float bval = (OP == 1 && colOK) ? bias[ecol] : 0.0f;
#pragma unroll
    for (int j = 0; j < 8; ++j) {
        float v = acc[j];
        if (OP == 1) {                          // dt = softplus(v + bias)
            float t = v + bval;
            v = (t > 20.0f) ? t : log1pf(__expf(t));
        }
        if (colOK) D[(size_t)(rbase + j) * ldD + ecol] = v;
    }
}

// ---------------------------------------------------------------------------
// Pass-1 input gather:  X1[n=(b*W+w), l=h, d] = x[b, h, w, d]
// ---------------------------------------------------------------------------
__global__ __launch_bounds__(256)
void pack_pass1(const float* __restrict__ x, float* __restrict__ X1)
{
    int i = blockIdx.x * 256 + threadIdx.x;
    if (i >= NTOK * D_MODEL) return;
    int d = i % D_MODEL;
    int t = i / D_MODEL;
    int l = t % SEQL;          // = h
    int n = t / SEQL;          // = b*56 + w
    int b = n / 56, w = n % 56;
    X1[i] = x[(((size_t)(b * 56 + l) * 56) + w) * D_MODEL + d];
}

// ---------------------------------------------------------------------------
// Inter-pass scrambled view (faithful to source):
//   X2[n, a, c] = O1[n, (a*192+c) % 56, (a*192+c) / 56]
// ---------------------------------------------------------------------------
__global__ __launch_bounds__(256)
void pack_pass2(const float* __restrict__ O1, float* __restrict__ X2)
{
    int i = blockIdx.x * 256 + threadIdx.x;
    if (i >= NTOK * D_MODEL) return;
    int c = i % D_MODEL;
    int t = i / D_MODEL;
    int a = t % SEQL;
    int n = t / SEQL;
    int q = a * D_MODEL + c;
    int hidx = q % 56;
    int didx = q / 56;
    X2[i] = O1[((size_t)(n * SEQL + hidx)) * D_MODEL + didx];
}

// ---------------------------------------------------------------------------
// Causal depthwise conv1d (+SiLU) for both scan directions.
// xi = first 384 columns of xz.  dir=1 processes the flipped sequence.
//   xc[dir,n,l,c] = silu(cb[c] + sum_k cw[c,k] * xi[n, src(l+k-2), c])
// ---------------------------------------------------------------------------
__global__ __launch_bounds__(256)
void conv_silu(const float* __restrict__ xz, const float* __restrict__ cw,
               const float* __restrict__ cb, float* __restrict__ xc)
{
    int i = blockIdx.x * 256 + threadIdx.x;
    if (i >= 2 * NTOK * D_INNER) return;
    int c   = i % D_INNER;
    int t   = i / D_INNER;
    int l   = t % SEQL;
    int t2  = t / SEQL;
    int n   = t2 % NSEQ;
    int dir = t2 / NSEQ;

    float s = cb[c];
#pragma unroll
    for (int k = 0; k < 3; ++k) {
        int j = l + k - 2;                     // position in dir-frame
        if (j >= 0) {
            int ls = dir ? (SEQL - 1 - j) : j; // source position in xz
            s += cw[c * 3 + k] * xz[((size_t)(n * SEQL + ls)) * 768 + c];
        }
    }
    xc[i] = s / (1.0f + __expf(-s));           // silu = s * sigmoid(s)
}

// ---------------------------------------------------------------------------
// Selective scan + C-contraction + D-skip + silu(z) gate, both directions
// summed (no back-flip, per source).  One thread owns one (seq, channel),
// keeping the 16-element state vector entirely in VGPRs.
// ---------------------------------------------------------------------------
__global__ __launch_bounds__(256)
void scan_gate(const float* __restrict__ xz,  const float* __restrict__ xc2,
               const float* __restrict__ dbl2, const float* __restrict__ dt2,
               const float* __restrict__ A_log, const float* __restrict__ Dskip,
               float* __restrict__ Y)
{
    int i = blockIdx.x * 256 + threadIdx.x;
    if (i >= NSEQ * D_INNER) return;
    int c = i % D_INNER;
    int n = i / D_INNER;

    float a[D_STATE];
#pragma unroll
    for (int j = 0; j < D_STATE; ++j) a[j] = -__expf(A_log[c * D_STATE + j]);
    float dsk = Dskip[c];

    for (int dir = 0; dir < 2; ++dir) {
        const float* xc  = xc2  + (size_t)dir * NTOK * D_INNER;
        const float* dbl = dbl2 + (size_t)dir * NTOK * XPROJ_N;
        const float* dt  = dt2  + (size_t)dir * NTOK * D_INNER;
        float h[D_STATE];
#pragma unroll
        for (int j = 0; j < D_STATE; ++j) h[j] = 0.0f;

        for (int l = 0; l < SEQL; ++l) {
            size_t base = (size_t)n * SEQL + l;
            float dtv = dt[base * D_INNER + c];
            float xcv = xc[base * D_INNER + c];
            const float* bc = dbl + base * XPROJ_N;
            float y = 0.0f;
#pragma unroll
            for (int j = 0; j < D_STATE; ++j) {
                h[j] = __expf(dtv * a[j]) * h[j] + dtv * bc[DT_RANK + j] * xcv;
                y += h[j] * bc[DT_RANK + D_STATE + j];
            }
            y += dsk * xcv;
            int ls = dir ? (SEQL - 1 - l) : l;
            float zv = xz[((size_t)n * SEQL + ls) * 768 + D_INNER + c];
            y *= zv / (1.0f + __expf(-zv));
            float* out = &Y[base * D_INNER + c];
            if (dir == 0) *out = y; else *out += y;
        }
    }
}

// ---------------------------------------------------------------------------
// Host-side launch helpers
// ---------------------------------------------------------------------------
template <bool MASKED, int OP>
static inline void launch_gemm(const float* A, int ldA, const float* W, int ldW,
                               int Nreal, const float* bias,
                               float* D, int ldD, int tilesM, int tilesN, int K,
                               hipStream_t s)
{
    int waves = tilesM * tilesN;
    int threads = waves * 32;
    int blocks = (threads + 255) / 256;
    wmma_gemm_f32<MASKED, OP><<<blocks, 256, 0, s>>>(A, ldA, W, ldW, Nreal,
                                                     bias, D, ldD,
                                                     tilesM, tilesN, K);
}

extern "C" void kernel_launch(void* const* d_in, const int* in_sizes, int n_in,
                              void* d_out, int out_size, void* d_ws, size_t ws_size,
                              hipStream_t stream)
{
    (void)in_sizes; (void)n_in; (void)out_size; (void)ws_size;
    const float* x = (const float*)d_in[0];
    // parameter sets: pass 0 -> h_*, pass 1 -> w_*
    const float* P[2][9];
    for (int s = 0; s < 2; ++s)
        for (int j = 0; j < 9; ++j)
            P[s][j] = (const float*)d_in[1 + s * 9 + j];

    float* ws = (float*)d_ws;
    size_t off = 0;
    float* X    = ws + off; off += (size_t)NTOK * D_MODEL;       // 1.20M
    float* xz   = ws + off; off += (size_t)NTOK * 768;           // 4.82M
    float* xc2  = ws + off; off += (size_t)2 * NTOK * D_INNER;   // 4.82M
    float* dbl2 = ws + off; off += (size_t)2 * NTOK * XPROJ_N;   // 0.55M
    float* dt2  = ws + off; off += (size_t)2 * NTOK * D_INNER;   // 4.82M
    float* Y    = ws + off; off += (size_t)NTOK * D_INNER;       // 2.41M
    float* O1   = ws + off; off += (size_t)NTOK * D_MODEL;       // 1.20M

    const int tilesM = NTOK / 16;                                // 392
    const int packBlocks = (NTOK * D_MODEL + 255) / 256;
    const int convBlocks = (2 * NTOK * D_INNER + 255) / 256;
    const int scanBlocks = (NSEQ * D_INNER + 255) / 256;

    for (int pass = 0; pass < 2; ++pass) {
        const float* const* p = P[pass];
        if (pass == 0)
            pack_pass1<<<packBlocks, 256, 0, stream>>>(x, X);
        else
            pack_pass2<<<packBlocks, 256, 0, stream>>>(O1, X);

        // xz = X @ in_w^T   (6272 x 192) * (768 x 192)^T
        launch_gemm<false, 0>(X, D_MODEL, p[0], D_MODEL, 768, nullptr,
                              xz, 768, tilesM, 768 / 16, D_MODEL, stream);

        // depthwise conv + silu, both directions
        conv_silu<<<convBlocks, 256, 0, stream>>>(xz, p[1], p[2], xc2);

        // dbl = xc @ xp_w^T  (N=44 padded to 3 tiles), per direction
        for (int dir = 0; dir < 2; ++dir)
            launch_gemm<true, 0>(xc2 + (size_t)dir * NTOK * D_INNER, D_INNER,
                                 p[3], D_INNER, XPROJ_N, nullptr,
                                 dbl2 + (size_t)dir * NTOK * XPROJ_N, XPROJ_N,
                                 tilesM, 3, D_INNER, stream);

        // dt = softplus(dbl[:, :12] @ dt_w^T + dt_b), per direction
        for (int dir = 0; dir < 2; ++dir)
            launch_gemm<false, 1>(dbl2 + (size_t)dir * NTOK * XPROJ_N, XPROJ_N,
                                  p[4], DT_RANK, D_INNER, p[5],
                                  dt2 + (size_t)dir * NTOK * D_INNER, D_INNER,
                                  tilesM, D_INNER / 16, DT_RANK, stream);

        // selective scan + gate, fwd+bwd summed
        scan_gate<<<scanBlocks, 256, 0, stream>>>(xz, xc2, dbl2, dt2,
                                                  p[6], p[7], Y);

        // O = Y @ out_w^T   (6272 x 384) * (192 x 384)^T
        float* Od = (pass == 0) ? O1 : (float*)d_out;
        launch_gemm<false, 0>(Y, D_INNER, p[8], D_INNER, D_MODEL, nullptr,
                              Od, D_MODEL, tilesM, D_MODEL / 16, D_INNER, stream);
    }
}